// GaussianSplatRenderer3D_52544629899273
// MI455X (gfx1250) — compile-verified
//
#include <hip/hip_runtime.h>

typedef __attribute__((ext_vector_type(2))) float v2f;
typedef __attribute__((ext_vector_type(8))) float v8f;

#define KRAD 5
#define KDIM 11

// ---------------------------------------------------------------------------
// Kernel 1: zero the [B*H*W, 5] accumulator in workspace (H,W read on-device).
// ---------------------------------------------------------------------------
__global__ __launch_bounds__(256) void zero_ws_kernel(float* __restrict__ ws,
                                                      const int* __restrict__ Hp,
                                                      const int* __restrict__ Wp,
                                                      int B) {
  const long n = (long)B * (*Hp) * (*Wp) * 5;
  const long stride = (long)gridDim.x * blockDim.x;
  for (long i = (long)blockIdx.x * blockDim.x + threadIdx.x; i < n; i += stride)
    ws[i] = 0.0f;
}

// ---------------------------------------------------------------------------
// Kernel 2: per-point splat. Camera transform Xc = R*pos + t done with
// v_wmma_f32_16x16x4_f32 (A = [R|t] rows, B = 16 point columns (x,y,z,1)).
// Each wave handles 32 points via two WMMAs; high half lanes recover their
// results from the second D matrix with a xor-16 shuffle. Then 11x11 taps
// scatter 5 atomic f32 adds each into the interleaved accumulator (L2-resident).
// ---------------------------------------------------------------------------
__global__ __launch_bounds__(256) void splat_kernel(
    const float* __restrict__ pos, const float* __restrict__ cov,
    const float* __restrict__ rgb, const float* __restrict__ opa,
    const float* __restrict__ Km, const float* __restrict__ Rm,
    const float* __restrict__ tv, float* __restrict__ acc,
    const int* __restrict__ Hp, const int* __restrict__ Wp,
    int N, long npts)
{
  const int H = *Hp, W = *Wp;
  const int lane = threadIdx.x & 31;
  const long wbase = (((long)blockIdx.x * blockDim.x) + threadIdx.x) & ~31L;
  const long p = wbase + lane;
  const bool valid = p < npts;
  const long pc = valid ? p : (npts - 1);
  const int b = (int)(pc / N);            // wave-uniform (N is a multiple of 32)
  const float* Rb = Rm + (long)b * 9;
  const float* Kb = Km + (long)b * 9;
  const float* tb = tv + (long)b * 3;

  const float x = pos[pc * 3 + 0];
  const float y = pos[pc * 3 + 1];
  const float zw = pos[pc * 3 + 2];

  float Xcx, Xcy, Xcz;
  if (wbase + 32 <= npts) {
    // Full wave: EXEC is all ones -> WMMA is legal.
    const bool lo = lane < 16;
    const int r = lane & 15;
    // A (16x4): lanes 0-15 = {K0,K1} of row r; lanes 16-31 = {K2,K3} of row r.
    v2f A;
    A[0] = (r < 3) ? (lo ? Rb[r * 3 + 0] : Rb[r * 3 + 2]) : 0.0f;
    A[1] = (r < 3) ? (lo ? Rb[r * 3 + 1] : tb[r]) : 0.0f;
    // B (4x16): VGPR0 = rows {0,2}, VGPR1 = rows {1,3}; columns = points.
    const float xo = __shfl_xor(x, 16, 32);
    const float yo = __shfl_xor(y, 16, 32);
    const float zo = __shfl_xor(zw, 16, 32);
    v2f B1, B2;
    B1[0] = lo ? x : zo;   B1[1] = lo ? y : 1.0f;   // points 0..15
    B2[0] = lo ? xo : zw;  B2[1] = lo ? yo : 1.0f;  // points 16..31
    v8f cz = {};
    // 8 args: (neg_a, A, neg_b, B, c_mod, C, reuse_a, reuse_b)
    v8f D1 = __builtin_amdgcn_wmma_f32_16x16x4_f32(
        false, A, false, B1, (short)0, cz, false, false);
    v8f D2 = __builtin_amdgcn_wmma_f32_16x16x4_f32(
        false, A, false, B2, (short)0, cz, false, false);
    // Lane n (<16): D1[r] = Xc component r of point n.
    // Lane 16+n: pull point (16+n)'s result out of D2 lanes 0..15.
    const float d2x = __shfl_xor(D2[0], 16, 32);
    const float d2y = __shfl_xor(D2[1], 16, 32);
    const float d2z = __shfl_xor(D2[2], 16, 32);
    Xcx = lo ? D1[0] : d2x;
    Xcy = lo ? D1[1] : d2y;
    Xcz = lo ? D1[2] : d2z;
  } else {
    // Partial tail wave: scalar fallback (keeps EXEC rule intact).
    Xcx = Rb[0] * x + Rb[1] * y + Rb[2] * zw + tb[0];
    Xcy = Rb[3] * x + Rb[4] * y + Rb[5] * zw + tb[1];
    Xcz = Rb[6] * x + Rb[7] * y + Rb[8] * zw + tb[2];
  }
  if (!valid) return;

  const float z = Xcz;
  const float invz = 1.0f / fmaxf(z, 1e-6f);
  const float xp = Xcx * invz, yp = Xcy * invz;
  const float K00 = Kb[0], K01 = Kb[1], K02 = Kb[2];
  const float K10 = Kb[3], K11 = Kb[4], K12 = Kb[5];
  const float u = K00 * xp + K01 * yp + K02;
  const float v = K10 * xp + K11 * yp + K12;
  const float sx = sqrtf(fmaxf(cov[pc * 3 + 0], 1e-9f)) * K00 * invz;
  const float sy = sqrtf(fmaxf(cov[pc * 3 + 1], 1e-9f)) * K11 * invz;
  const float isx = 1.0f / fmaxf(sx, 1e-6f);
  const float isy = 1.0f / fmaxf(sy, 1e-6f);
  const float cr = rgb[pc * 3 + 0], cg = rgb[pc * 3 + 1], cb = rgb[pc * 3 + 2];
  const float op = opa[pc];

  // Separable Gaussian: 22 exps instead of 121.
  float ex[KDIM], ey[KDIM];
#pragma unroll
  for (int i = 0; i < KDIM; ++i) {
    const float o = (float)(i - KRAD);
    const float gx = o * isx, gy = o * isy;
    ex[i] = __expf(-0.5f * gx * gx);
    ey[i] = __expf(-0.5f * gy * gy) * op;
  }

  float* __restrict__ accb = acc + (long)b * H * W * 5;
  for (int iy = 0; iy < KDIM; ++iy) {
    const float pyf = rintf(v + (float)(iy - KRAD) * sy);  // round-half-even
    if (!(pyf >= 0.0f && pyf <= (float)(H - 1))) continue;
    const long rowoff = (long)((int)pyf) * W;
    const float wy = ey[iy];
#pragma unroll
    for (int ix = 0; ix < KDIM; ++ix) {
      const float pxf = rintf(u + (float)(ix - KRAD) * sx);
      if (!(pxf >= 0.0f && pxf <= (float)(W - 1))) continue;
      const float alpha = wy * ex[ix];
      if (alpha == 0.0f) continue;     // adding 0 is a no-op; save the atomic
      float* a5 = accb + (rowoff + (int)pxf) * 5;
      atomicAdd(a5 + 0, alpha * cr);
      atomicAdd(a5 + 1, alpha * cg);
      atomicAdd(a5 + 2, alpha * cb);
      atomicAdd(a5 + 3, alpha);
      atomicAdd(a5 + 4, alpha * z);
    }
  }
}

// ---------------------------------------------------------------------------
// Kernel 3: normalize and write planar NCHW outputs:
//   rgb_img [B,3,H,W] followed by depth_img [B,1,H,W] in d_out.
// ---------------------------------------------------------------------------
__global__ __launch_bounds__(256) void finalize_kernel(
    const float* __restrict__ acc, float* __restrict__ out,
    const int* __restrict__ Hp, const int* __restrict__ Wp, int B)
{
  const long HW = (long)(*Hp) * (*Wp);
  const long total = (long)B * HW;
  const long stride = (long)gridDim.x * blockDim.x;
  for (long i = (long)blockIdx.x * blockDim.x + threadIdx.x; i < total; i += stride) {
    const float* a5 = acc + i * 5;
    const float inv = 1.0f / fmaxf(a5[3], 1e-6f);
    const long b = i / HW;
    const long pix = i - b * HW;
    float* rgbp = out + b * 3 * HW + pix;
    rgbp[0]      = a5[0] * inv;
    rgbp[HW]     = a5[1] * inv;
    rgbp[2 * HW] = a5[2] * inv;
    out[(long)B * 3 * HW + b * HW + pix] = a5[4] * inv;
  }
}

// ---------------------------------------------------------------------------
extern "C" void kernel_launch(void* const* d_in, const int* in_sizes, int n_in,
                              void* d_out, int out_size, void* d_ws, size_t ws_size,
                              hipStream_t stream) {
  const float* pos = (const float*)d_in[0];
  const float* cov = (const float*)d_in[1];
  const float* rgb = (const float*)d_in[2];
  const float* opa = (const float*)d_in[3];
  const float* Km  = (const float*)d_in[4];
  const float* Rm  = (const float*)d_in[5];
  const float* tv  = (const float*)d_in[6];
  const int* Hp    = (const int*)d_in[7];
  const int* Wp    = (const int*)d_in[8];

  float* acc = (float*)d_ws;   // [B*H*W, 5] interleaved accumulator (~21 MB)
  float* out = (float*)d_out;

  const int B = in_sizes[4] / 9;            // K is [B,3,3]
  const long npts = (long)in_sizes[0] / 3;  // B*N
  const int N = (int)(npts / B);

  const dim3 blk(256);
  zero_ws_kernel<<<2048, blk, 0, stream>>>(acc, Hp, Wp, B);
  const unsigned sblocks = (unsigned)((npts + 255) / 256);
  splat_kernel<<<sblocks, blk, 0, stream>>>(pos, cov, rgb, opa, Km, Rm, tv,
                                            acc, Hp, Wp, N, npts);
  finalize_kernel<<<2048, blk, 0, stream>>>(acc, out, Hp, Wp, B);
}